// Cross_IAN_62208306315584
// MI455X (gfx1250) — compile-verified
//
#include <hip/hip_runtime.h>

typedef __attribute__((ext_vector_type(2))) float v2f;
typedef __attribute__((ext_vector_type(8))) float v8f;

#define B_DIM 64
#define S_TOT 1024
#define D_DIM 768
#define SSPLIT 4
#define SCHUNK (S_TOT / SSPLIT)   // 256 s-rows per block
#define ITERS  (SCHUNK / 4)       // 64 WMMA steps (K=4 each)

// out[b,d] = (1/S) * sum_s x0[b,s,d], computed as a ones-matrix WMMA reduction.
// Each wave: 16-column d-tile, A = all (1/1024), B = 4 rows of x0 per step.
__global__ __launch_bounds__(256) void mean_rows_wmma(const float* __restrict__ x0,
                                                      float* __restrict__ out) {
    const int lane = threadIdx.x & 31;
    const int wave = threadIdx.x >> 5;
    const int n    = lane & 15;   // column within 16-wide tile (B/C/D lane mapping)
    const int h    = lane >> 4;   // K half selector (any bijection over K is fine for a sum)

    const int b      = blockIdx.y;
    const int d_base = blockIdx.x * 128 + wave * 16;
    const int s0     = blockIdx.z * SCHUNK;

    const float* p = x0 + ((size_t)b * S_TOT + (size_t)(s0 + h)) * (size_t)D_DIM
                        + (size_t)(d_base + n);

    v2f aones;
    aones.x = 1.0f / (float)S_TOT;
    aones.y = 1.0f / (float)S_TOT;

    v8f acc0 = {};
    v8f acc1 = {};

    // Each pair of steps consumes 8 s-rows; pointer walks s in units of 8*D.
#pragma unroll 4
    for (int it = 0; it < ITERS; it += 2) {
        v2f bm0, bm1;
        bm0.x = p[0];               // s0+4*it + h
        bm0.y = p[2 * D_DIM];       // s0+4*it + 2 + h
        bm1.x = p[4 * D_DIM];       // s0+4*it + 4 + h
        bm1.y = p[6 * D_DIM];       // s0+4*it + 6 + h
        acc0 = __builtin_amdgcn_wmma_f32_16x16x4_f32(false, aones, false, bm0,
                                                     (short)0, acc0, false, false);
        acc1 = __builtin_amdgcn_wmma_f32_16x16x4_f32(false, aones, false, bm1,
                                                     (short)0, acc1, false, false);
        p += 8 * D_DIM;
    }

    // D[0, n] lives in acc element 0 of lanes 0..15 (all M rows identical).
    if (lane < 16) {
        atomicAdd(&out[(size_t)b * D_DIM + (size_t)(d_base + n)], acc0[0] + acc1[0]);
    }
}

extern "C" void kernel_launch(void* const* d_in, const int* in_sizes, int n_in,
                              void* d_out, int out_size, void* d_ws, size_t ws_size,
                              hipStream_t stream) {
    const float* x0 = (const float*)d_in[0];   // [B,S,D] fp32; x1/W1/W2 are dead
    float* out = (float*)d_out;                // [B,D] fp32

    hipMemsetAsync(d_out, 0, (size_t)out_size * sizeof(float), stream);

    dim3 grid(D_DIM / 128, B_DIM, SSPLIT);     // 6 x 64 x 4 = 1536 blocks, 8 waves each
    mean_rows_wmma<<<grid, 256, 0, stream>>>(x0, out);
}